// YOLOLoss_82042465289124
// MI455X (gfx1250) — compile-verified
//
#include <hip/hip_runtime.h>
#include <math.h>

// Problem constants (fixed by reference setup_inputs)
#define BS      32
#define TPB     50            // targets per image
#define NA      3             // anchors
#define NCLS    20            // NUM_CLASSES-1 class channels
#define H       104
#define W       104
#define HW      (H*W)         // 10816
#define CH      75            // NA*(4+21)
#define NCELLS  (BS*NA*HW)    // 1,038,336
#define HW4     (HW/4)        // 2704

typedef __attribute__((ext_vector_type(2))) float v2f;
typedef __attribute__((ext_vector_type(8))) float v8f;

// min(log(1+e^z), 100) == -clip(log(1-sigmoid(z)), -100) == -clip(log(sigmoid(-z)), -100)
__device__ __forceinline__ float sp100(float z) {
    float r = (z > 15.0f) ? z : log1pf(expf(z));
    return fminf(r, 100.0f);
}

// Decode one target: validity (incl. index-drop range check), cell, best anchor, IoUs.
__device__ __forceinline__ void decode_target(const float* __restrict__ tgt,
                                              int b, int tt,
                                              bool& valid, int& gi, int& gj, int& best,
                                              float iou[NA],
                                              float& gx, float& gy, float& gw, float& gh) {
    const float* t = tgt + ((size_t)b * TPB + tt) * 5;
    float t0 = t[0], t1 = t[1], t2 = t[2], t3 = t[3], t4 = t[4];
    valid = (t0 + t1 + t2 + t3 + t4) > 0.0f;
    gx = t0 * W; gy = t1 * H; gw = t2 * W; gh = t3 * H;
    gi = (int)gx; gj = (int)gy;
    if (gi < 0 || gi >= W || gj < 0 || gj >= H) valid = false;   // mode='drop'
    const float aw[NA] = {14.5f, 19.5f, 46.625f};                // ANCHORS / stride(8)
    const float ah[NA] = {11.25f, 24.75f, 40.75f};
    float area_gt = (gw + 1.0f) * (gh + 1.0f);
    best = 0;
    float bestv = -1.0f;
    for (int a = 0; a < NA; a++) {
        float iw = fmaxf(fminf(gw, aw[a]) + 1.0f, 0.0f);
        float ih = fmaxf(fminf(gh, ah[a]) + 1.0f, 0.0f);
        float inter = iw * ih;
        float uni = area_gt + (aw[a] + 1.0f) * (ah[a] + 1.0f) - inter + 1e-16f;
        iou[a] = inter / uni;
        if (iou[a] > bestv) { bestv = iou[a]; best = a; }        // first-max wins
    }
}

// acc slots: 0:x 1:y 2:w 3:h 4:obj_mask 5:obj_dense(+corr) 6:cls 7:npos
__global__ void init_acc(float* __restrict__ acc) {
    if (threadIdx.x < 8) acc[threadIdx.x] = 0.0f;
}

// Dense term: sum over all (b,a,j,i) of min(softplus(conf_logit), 100).
// Only the 3 conf channels are read (4.15 MB, float4/B128 coalesced).
// Reduction: per-wave via V_WMMA_F32_16X16X4_F32 (A = ones -> column sums),
// then 4 shfl_xor steps, then LDS across waves -> 1 atomic per block (256 total).
__global__ void dense_conf(const float* __restrict__ inp, float* __restrict__ acc) {
    float s = 0.0f;
    const int nchunks = NCELLS / 4;
    for (int idx = blockIdx.x * blockDim.x + threadIdx.x; idx < nchunks;
         idx += gridDim.x * blockDim.x) {
        int p4 = idx % HW4;
        int ba = idx / HW4;
        int a  = ba % NA;
        int b  = ba / NA;
        const float4* ptr =
            (const float4*)(inp + (size_t)(b * CH + a * 25 + 4) * HW) + p4;
        float4 v = *ptr;
        s += sp100(v.x) + sp100(v.y) + sp100(v.z) + sp100(v.w);
    }

    // Wave32 reduce through the matrix unit (EXEC is all-ones here: no divergence).
    float wave_total;
#if __has_builtin(__builtin_amdgcn_wmma_f32_16x16x4_f32)
    {
        v2f aones = {1.0f, 1.0f};
        v2f bvals = {s, s};                    // 64 B-slots carry each partial twice
        v8f c = {};
        c = __builtin_amdgcn_wmma_f32_16x16x4_f32(
                false, aones, false, bvals, (short)0, c, false, false);
        float colsum = c[0];                   // row M=0 (lanes 0-15) / M=8 (16-31): identical
        colsum += __shfl_xor(colsum, 1, 32);
        colsum += __shfl_xor(colsum, 2, 32);
        colsum += __shfl_xor(colsum, 4, 32);
        colsum += __shfl_xor(colsum, 8, 32);
        wave_total = colsum * 0.5f;            // undo the x2 duplication in B
    }
#else
    {
        wave_total = s;
        for (int m = 1; m < 32; m <<= 1) wave_total += __shfl_xor(wave_total, m, 32);
    }
#endif

    __shared__ float wsum[8];                  // 256 threads = 8 waves
    int wave = threadIdx.x >> 5;
    if ((threadIdx.x & 31) == 0) wsum[wave] = wave_total;
    __syncthreads();
    if (threadIdx.x == 0) {
        float bt = 0.0f;
        for (int i = 0; i < 8; i++) bt += wsum[i];
        atomicAdd(&acc[5], bt);
    }
}

// Sparse part: one thread per (b, t). Duplicate scatters can only come from
// targets of the same image, so dedup by rescanning earlier targets (t' < t)
// instead of claim arrays — removes 8.3 MB of workspace init + CAS traffic.
__global__ void target_kernel(const float* __restrict__ inp,
                              const float* __restrict__ tgt,
                              float* __restrict__ acc) {
    int tid = blockIdx.x * blockDim.x + threadIdx.x;
    if (tid >= BS * TPB) return;
    int b = tid / TPB, tt = tid % TPB;

    bool valid; int gi, gj, best; float iou[NA], gx, gy, gw, gh;
    decode_target(tgt, b, tt, valid, gi, gj, best, iou, gx, gy, gw, gh);
    if (!valid) return;

    bool winMask = true;
    bool winNo[NA] = {iou[0] > 0.5f, iou[1] > 0.5f, iou[2] > 0.5f};
    for (int u = 0; u < tt; u++) {
        bool v2; int gi2, gj2, b2; float iou2[NA], d0, d1, d2, d3;
        decode_target(tgt, b, u, v2, gi2, gj2, b2, iou2, d0, d1, d2, d3);
        if (!v2 || gi2 != gi || gj2 != gj) continue;
        if (b2 == best) winMask = false;
        for (int a = 0; a < NA; a++)
            if (iou2[a] > 0.5f) winNo[a] = false;
    }

    // noobj corrections: remove dense contribution where noobj got zeroed
    for (int a = 0; a < NA; a++) {
        if (winNo[a]) {
            float z = inp[(size_t)(b * CH + a * 25 + 4) * HW + gj * W + gi];
            atomicAdd(&acc[5], -sp100(z));
        }
    }

    // mask cell: gather the 25 prediction channels, accumulate loss terms
    if (winMask) {
        const float aw[NA] = {14.5f, 19.5f, 46.625f};
        const float ah[NA] = {11.25f, 24.75f, 40.75f};
        size_t base = (size_t)(b * CH + best * 25) * HW + (size_t)gj * W + gi;
        float px = inp[base];
        float py = inp[base + (size_t)1 * HW];
        float pw = inp[base + (size_t)2 * HW];
        float ph = inp[base + (size_t)3 * HW];
        float pc = inp[base + (size_t)4 * HW];

        float txv = gx - (float)gi;
        float tyv = gy - (float)gj;
        float twv = logf(gw / aw[best] + 1e-16f);
        float thv = logf(gh / ah[best] + 1e-16f);

        // bce(sigmoid(z), t) = t*min(sp(-z),100) + (1-t)*min(sp(z),100)
        atomicAdd(&acc[0], txv * sp100(-px) + (1.0f - txv) * sp100(px));
        atomicAdd(&acc[1], tyv * sp100(-py) + (1.0f - tyv) * sp100(py));
        atomicAdd(&acc[2], (pw - twv) * (pw - twv));
        atomicAdd(&acc[3], (ph - thv) * (ph - thv));
        atomicAdd(&acc[4], sp100(-pc));                      // bce(conf, 1)

        int ci = (int)(tgt[((size_t)b * TPB + tt) * 5 + 4]);
        float cls_sum = 0.0f;
        for (int c = 0; c < NCLS; c++) {
            float z = inp[base + (size_t)(5 + c) * HW];
            cls_sum += (c == ci) ? sp100(-z) : sp100(z);
        }
        atomicAdd(&acc[6], cls_sum);
        atomicAdd(&acc[7], 1.0f);                            // npos
    }
}

__global__ void finalize_kernel(const float* __restrict__ acc,
                                float* __restrict__ out) {
    if (blockIdx.x == 0 && threadIdx.x == 0) {
        const float N = (float)NCELLS;
        out[0] = acc[0] / N * 2.5f;                          // loss_x
        out[1] = acc[1] / N * 2.5f;                          // loss_y
        out[2] = acc[2] / N * 2.5f;                          // loss_w
        out[3] = acc[3] / N * 2.5f;                          // loss_h
        out[4] = acc[4] / N + 0.5f * acc[5] / N;             // loss_obj (L_OBJ=1)
        out[5] = acc[6] / fmaxf(acc[7] * (float)NCLS, 1.0f); // loss_cls (L_CLS=1)
    }
}

extern "C" void kernel_launch(void* const* d_in, const int* in_sizes, int n_in,
                              void* d_out, int out_size, void* d_ws, size_t ws_size,
                              hipStream_t stream) {
    const float* inp = (const float*)d_in[0];   // (32, 75, 104, 104) fp32
    const float* tgt = (const float*)d_in[1];   // (32, 50, 5) fp32
    float* out = (float*)d_out;                 // 6 fp32 losses
    float* acc = (float*)d_ws;                  // 8 floats of scratch

    init_acc<<<1, 32, 0, stream>>>(acc);
    dense_conf<<<256, 256, 0, stream>>>(inp, acc);           // 256 atomics total
    target_kernel<<<(BS * TPB + 31) / 32, 32, 0, stream>>>(inp, tgt, acc);
    finalize_kernel<<<1, 32, 0, stream>>>(acc, out);
}